// StripePolynomial2d_59846074302911
// MI455X (gfx1250) — compile-verified
//
#include <hip/hip_runtime.h>

// ---------------------------------------------------------------------------
// StripePolynomial2d for MI455X (gfx1250)
//
// Strategy:
//  1) Prep kernel: convert per-segment Lagrange coefficients (nodes -1,-.5,.5,1)
//     to monomial (Horner) coefficients with V_WMMA_F32_16X16X4_F32 (full fp32
//     matrix op -> no precision loss). m[rot][ch][seg] = float4 -> d_ws (24 KB).
//  2) Main kernel: bandwidth-bound streaming pass. m cached in LDS (24 KB),
//     per element per rotation: segment id + local t, one ds_load_b128 gather,
//     3-FMA Horner. Position maps computed analytically per pixel (double,
//     matching the reference's float64 precompute) and reused across B*C=48
//     elements per thread.
// ---------------------------------------------------------------------------

typedef float v2f __attribute__((ext_vector_type(2)));
typedef float v8f __attribute__((ext_vector_type(8)));

#define NPTS      4
#define SEGMENTS  128
#define NCOEFF    385      // SEGMENTS*(NPTS-1)+1
#define ROTS      4
#define CHANS     3
#define IMG_H     512
#define IMG_W     512
#define BATCH     16

// Lagrange(-1,-0.5,0.5,1) -> monomial conversion: row k = coefficient of t^k,
// column j = Lagrange basis index.
__device__ __constant__ float c_T[4][4] = {
    {-1.0f/6.0f,  2.0f/3.0f,  2.0f/3.0f, -1.0f/6.0f},   // t^0
    { 1.0f/6.0f, -4.0f/3.0f,  4.0f/3.0f, -1.0f/6.0f},   // t^1
    { 2.0f/3.0f, -2.0f/3.0f, -2.0f/3.0f,  2.0f/3.0f},   // t^2
    {-2.0f/3.0f,  4.0f/3.0f, -4.0f/3.0f,  2.0f/3.0f},   // t^3
};

// ---------------------------------------------------------------------------
// Kernel 1: coefficient prep via fp32 WMMA.
//   grid.x = ROTS*CHANS (12), block = 32 (one wave; EXEC all-ones for WMMA).
//   D(16x16) = A(16x4: T padded with zero rows) * B(4x16: 16 segments' coeffs)
//   Rows 0..3 of D are m0..m3 for 16 consecutive segments.
// ---------------------------------------------------------------------------
__global__ __launch_bounds__(32)
void stripe_coeff_prep(const float* __restrict__ w, float* __restrict__ m_out)
{
    const int rc   = blockIdx.x;          // rot*CHANS + ch
    const int lane = threadIdx.x;         // 0..31
    const int row  = lane & 15;
    const bool hi  = (lane >= 16);        // lanes 16..31 hold K=2,3

    // A-matrix (16x4 fp32): lane<16 -> A[row][0..1], lane>=16 -> A[row][2..3]
    v2f a;
    if (row < 4) {
        a.x = c_T[row][hi ? 2 : 0];
        a.y = c_T[row][hi ? 3 : 1];
    } else {
        a.x = 0.0f; a.y = 0.0f;
    }

    const float* wb = w + (size_t)rc * NCOEFF;

    for (int sb = 0; sb < SEGMENTS / 16; ++sb) {
        const int seg = sb * 16 + row;
        // B-matrix (4x16): column n = segment (sb*16+n), row j = coeff j.
        // lane<16 holds B[0..1][n], lane>=16 holds B[2..3][n].
        const int off = 3 * seg + (hi ? 2 : 0);
        v2f b;
        b.x = wb[off];
        b.y = wb[off + 1];

        v8f acc = {};
        acc = __builtin_amdgcn_wmma_f32_16x16x4_f32(
                  /*neg_a=*/false, a, /*neg_b=*/false, b,
                  /*c_mod=*/(short)0, acc, /*reuse_a=*/false, /*reuse_b=*/false);

        // D rows 0..7 are in lanes 0..15 (vgpr r -> row r). We need rows 0..3.
        if (!hi) {
            float4 mv = make_float4(acc[0], acc[1], acc[2], acc[3]);
            ((float4*)m_out)[(size_t)rc * SEGMENTS + seg] = mv;
        }
    }
}

// ---------------------------------------------------------------------------
// Kernel 2: streaming evaluation. Block = 256 threads along W; each thread
// owns one (h,w) pixel and loops over (b,c), reusing the 4 position values.
// ---------------------------------------------------------------------------
__global__ __launch_bounds__(256)
void stripe_poly_eval(const float* __restrict__ x,
                      const float* __restrict__ m_in,
                      float* __restrict__ out)
{
    __shared__ float4 mlds[ROTS * CHANS * SEGMENTS];   // 1536 * 16B = 24 KB

    const int tid = threadIdx.x;
    const float4* m4 = (const float4*)m_in;
#pragma unroll
    for (int i = 0; i < (ROTS * CHANS * SEGMENTS) / 256; ++i)
        mlds[tid + i * 256] = m4[tid + i * 256];
    __syncthreads();

    const int wi = blockIdx.x * 256 + tid;
    const int h  = blockIdx.y;

    // Position maps (rotation list interleaves sgn=+1,-1 per theta; forward
    // consumes entries 0..3 => theta in {0,0,pi/8,pi/8}, sgn {+,-,+,-}).
    // Computed in double to match the reference's float64 precompute; the
    // per-rotation min/max occur at grid corners, so they are exact.
    const double c8 = 0.9238795325112867;     // cos(pi/8)
    const double s8 = 0.3826834323650898;     // sin(pi/8)
    const double hd = (double)h, wd = (double)wi;

    float pos[ROTS];
    pos[0] = (float)((hd * (1.0 / 511.0) - 0.5) * 512.0);
    pos[1] = pos[0];                           // theta=0: sgn irrelevant
    const double dn = 1.0 / ((c8 + s8) * 511.0);
    pos[2] = (float)(((c8 * hd + s8 * wd) * dn - 0.5) * 512.0);
    pos[3] = (float)(((c8 * hd - s8 * wd + s8 * 511.0) * dn - 0.5) * 512.0);

    const float inv512 = 1.0f / 512.0f;

    for (int b = 0; b < BATCH; ++b) {
#pragma unroll
        for (int c = 0; c < CHANS; ++c) {
            const size_t idx =
                (((size_t)(b * CHANS + c) * IMG_H + h) * IMG_W) + wi;
            const float xv = x[idx];
            float acc = 0.0f;
#pragma unroll
            for (int r = 0; r < ROTS; ++r) {
                const float s = xv + pos[r];
                const float u = fmaf(s, inv512, 1.0f) * 64.0f; // (dl+1)*S/2
                float sidf = floorf(u);
                sidf = fminf(fmaxf(sidf, 0.0f), 127.0f);
                const float t = fmaf(2.0f, u - sidf, -1.0f);
                const float4 m = mlds[(r * CHANS + c) * SEGMENTS + (int)sidf];
                acc += fmaf(fmaf(fmaf(m.w, t, m.z), t, m.y), t, m.x);
            }
            out[idx] = acc;
        }
    }
}

// ---------------------------------------------------------------------------
extern "C" void kernel_launch(void* const* d_in, const int* in_sizes, int n_in,
                              void* d_out, int out_size, void* d_ws, size_t ws_size,
                              hipStream_t stream)
{
    const float* x = (const float*)d_in[0];   // [16,3,512,512] fp32
    const float* w = (const float*)d_in[1];   // [4,3,385] fp32
    float* out  = (float*)d_out;              // [16,3,512,512] fp32
    float* m_ws = (float*)d_ws;               // needs 4*3*128*4 floats = 24 KB

    (void)in_sizes; (void)n_in; (void)out_size; (void)ws_size;

    stripe_coeff_prep<<<dim3(ROTS * CHANS), dim3(32), 0, stream>>>(w, m_ws);
    stripe_poly_eval<<<dim3(IMG_W / 256, IMG_H), dim3(256), 0, stream>>>(x, m_ws, out);
}